// ActorCritic_70514773065814
// MI455X (gfx1250) — compile-verified
//
#include <hip/hip_runtime.h>
#include <hip/hip_bf16.h>

// ---------------------------------------------------------------------------
// Problem constants (from reference): B=32, L=256 -> M=8192 tokens,
// IN_DIM=256, D=512, N=4, S=8 sublayers, Sinkhorn 100 iters.
// ---------------------------------------------------------------------------
#define MTOK   8192
#define DDIM   512
#define INDIM  256
#define NSTR   4
#define NSUB   8
#define NC     2048      // N*D
#define NCOLS  24        // N*N + 2N
#define SK_IT  100

#define LDS_STRIDE 36    // 32 + 4 pad: keeps 16B row alignment, conflict-free frag reads

typedef float v2f __attribute__((ext_vector_type(2)));
typedef float v8f __attribute__((ext_vector_type(8)));

__device__ __forceinline__ float sigmoidf_(float x) {
    return 1.0f / (1.0f + __expf(-x));
}

// Low 32 bits of a generic pointer to __shared__ = wave-relative LDS byte offset
__device__ __forceinline__ unsigned lds_off(const void* p) {
    return (unsigned)(uintptr_t)p;
}

__device__ __forceinline__ void async_copy_b128(unsigned lds_byte, const float* gptr) {
    asm volatile("global_load_async_to_lds_b128 %0, %1, off"
                 :: "v"(lds_byte), "v"((unsigned long long)(uintptr_t)gptr)
                 : "memory");
}

__device__ __forceinline__ void wait_async0() {
    asm volatile("s_wait_asynccnt 0x0" ::: "memory");
}

// ---------------------------------------------------------------------------
// f32 WMMA GEMM:  C[M,512] = A[M,Kdim] * W[512,Kdim]^T (+bias) (opt SiLU)
// Block tile: 64(M) x 128(N), 8 waves, each wave 16(M) x 64(N) = 4 frags.
// Double-buffered LDS staged via GLOBAL_LOAD_ASYNC_TO_LDS_B128 (ASYNCcnt),
// one barrier per K-chunk. Both tiles stored K-major so WMMA fragment reads
// are single aligned ds_load_b64 (no repacking).
// grid = (512/128, M/64). If bcast4: write to 4 replicated D-streams of h.
// ---------------------------------------------------------------------------
__global__ void __launch_bounds__(256)
gemm_wmma_f32(const float* __restrict__ A, const float* __restrict__ W,
              const float* __restrict__ bias, float* __restrict__ C,
              int Kdim, int act_silu, int bcast4) {
    __shared__ __align__(16) float Alds[2][64][LDS_STRIDE];   // [buf][m][k]
    __shared__ __align__(16) float Blds[2][128][LDS_STRIDE];  // [buf][n][k]

    const int tid  = threadIdx.x;
    const int lane = tid & 31;
    const int wid  = tid >> 5;
    const int mWave = (wid & 3) * 16;
    const int nWave = (wid >> 2) * 64;
    const int mBase = blockIdx.y * 64;
    const int nBase = blockIdx.x * 128;

    // Issue async copies for one 32-wide K chunk into LDS buffer `buf`.
    auto issue_tiles = [&](int kc, int buf) {
        // A tile: 64 x 32 floats = 512 float4; 2 per thread (16B coalesced)
        #pragma unroll
        for (int i = 0; i < 2; ++i) {
            int q = tid + i * 256;
            int m = q >> 3, k4 = q & 7;          // 8 float4 per 32-float row
            async_copy_b128(lds_off(&Alds[buf][m][k4 * 4]),
                            A + (size_t)(mBase + m) * Kdim + kc + k4 * 4);
        }
        // B tile (k-major, no transpose): 128 x 32 floats = 1024 float4; 4/thread
        #pragma unroll
        for (int i = 0; i < 4; ++i) {
            int q = tid + i * 256;
            int n = q >> 3, k4 = q & 7;
            async_copy_b128(lds_off(&Blds[buf][n][k4 * 4]),
                            W + (size_t)(nBase + n) * Kdim + kc + k4 * 4);
        }
    };

    v8f acc[4] = {};

    const int rowA  = mWave + (lane & 15);
    const int kOff  = (lane < 16) ? 0 : 2;   // A/B VGPR K-split per ISA layout
    const int colLo = lane & 15;
    const int nCh   = Kdim >> 5;

    issue_tiles(0, 0);

    for (int c = 0; c < nCh; ++c) {
        const int buf = c & 1;
        wait_async0();          // this wave's async copies (chunk c) landed
        __syncthreads();        // all waves' copies landed; compute c-1 done
        if (c + 1 < nCh) issue_tiles((c + 1) << 5, buf ^ 1);

        #pragma unroll
        for (int k = 0; k < 32; k += 4) {
            v2f a = *(const v2f*)&Alds[buf][rowA][k + kOff];
            #pragma unroll
            for (int ns = 0; ns < 4; ++ns) {
                int col = nWave + ns * 16 + colLo;
                v2f b = *(const v2f*)&Blds[buf][col][k + kOff];
                acc[ns] = __builtin_amdgcn_wmma_f32_16x16x4_f32(
                    false, a, false, b, (short)0, acc[ns], false, false);
            }
        }
    }

    // Epilogue: bias (+SiLU), write out. C/D layout: VGPR g -> rows g / g+8.
    const int rowSel = (lane < 16) ? 0 : 8;
    #pragma unroll
    for (int ns = 0; ns < 4; ++ns) {
        #pragma unroll
        for (int g = 0; g < 8; ++g) {
            int row = mBase + mWave + rowSel + g;
            int col = nBase + nWave + ns * 16 + colLo;
            float v = acc[ns][g] + bias[col];
            if (act_silu) v = v * sigmoidf_(v);
            if (bcast4) {
                float* o = C + (size_t)row * NC + col;
                o[0] = v; o[DDIM] = v; o[2 * DDIM] = v; o[3 * DDIM] = v;
            } else {
                C[(size_t)row * DDIM + col] = v;
            }
        }
    }
}

// ---------------------------------------------------------------------------
// Per-token stats: H[24] = (norm_w .* hf) @ w, r, gates, Sinkhorn -> coeffs.
// One wave per token; 8 waves / block.
// coeffs[t][0:4]=H_pre, [4:8]=H_post, [8:24]=P (row-major 4x4)
// ---------------------------------------------------------------------------
__global__ void __launch_bounds__(256)
stats_kernel(const float* __restrict__ h, const float* __restrict__ norm_w,
             const float* __restrict__ w, const float* __restrict__ alpha,
             const float* __restrict__ beta, float* __restrict__ coeffs) {
    const int lane = threadIdx.x & 31;
    const int t = blockIdx.x * 8 + (threadIdx.x >> 5);
    const float* hf = h + (size_t)t * NC;

    float H[NCOLS];
    #pragma unroll
    for (int j = 0; j < NCOLS; ++j) H[j] = 0.0f;
    float ssq = 0.0f;

    #pragma unroll 4
    for (int i = 0; i < NC / 32; ++i) {
        int c = i * 32 + lane;
        float hv = hf[c];
        ssq += hv * hv;
        float g = norm_w[c] * hv;
        const float4* wr = (const float4*)(w + (size_t)c * NCOLS);
        #pragma unroll
        for (int q = 0; q < 6; ++q) {
            float4 wv = wr[q];
            H[q * 4 + 0] += g * wv.x;
            H[q * 4 + 1] += g * wv.y;
            H[q * 4 + 2] += g * wv.z;
            H[q * 4 + 3] += g * wv.w;
        }
    }

    // Wave32 butterfly reduction (all lanes end with totals)
    #pragma unroll
    for (int off = 16; off > 0; off >>= 1) {
        ssq += __shfl_xor(ssq, off, 32);
        #pragma unroll
        for (int j = 0; j < NCOLS; ++j) H[j] += __shfl_xor(H[j], off, 32);
    }

    const float r  = sqrtf(ssq) * (1.0f / 45.254833995939045f);  // /sqrt(2048)
    const float r_ = 1.0f / (r + 1e-6f);
    const float a0 = alpha[0], a1 = alpha[1], a2 = alpha[2];

    float Hpre[4], Hpost[4], Amat[16];
    #pragma unroll
    for (int n = 0; n < 4; ++n) {
        Hpre[n]  = sigmoidf_(r_ * H[n] * a0 + beta[n]);
        Hpost[n] = 2.0f * sigmoidf_(r_ * H[4 + n] * a1 + beta[4 + n]);
    }
    #pragma unroll
    for (int e = 0; e < 16; ++e)
        Amat[e] = __expf(r_ * H[8 + e] * a2 + beta[8 + e]);

    // Sinkhorn-Knopp: u = 1/(A v + eps); v = 1/(A^T u + eps)
    float u[4] = {1.f, 1.f, 1.f, 1.f}, v[4] = {1.f, 1.f, 1.f, 1.f};
    #pragma unroll 1
    for (int it = 0; it < SK_IT; ++it) {
        float nu[4];
        #pragma unroll
        for (int m = 0; m < 4; ++m)
            nu[m] = 1.0f / (Amat[4 * m + 0] * v[0] + Amat[4 * m + 1] * v[1] +
                            Amat[4 * m + 2] * v[2] + Amat[4 * m + 3] * v[3] + 1e-8f);
        #pragma unroll
        for (int n = 0; n < 4; ++n)
            v[n] = 1.0f / (Amat[n] * nu[0] + Amat[4 + n] * nu[1] +
                           Amat[8 + n] * nu[2] + Amat[12 + n] * nu[3] + 1e-8f);
        #pragma unroll
        for (int m = 0; m < 4; ++m) u[m] = nu[m];
    }

    if (lane == 0) {
        float* o = coeffs + (size_t)t * NCOLS;
        #pragma unroll
        for (int n = 0; n < 4; ++n) { o[n] = Hpre[n]; o[4 + n] = Hpost[n]; }
        #pragma unroll
        for (int m = 0; m < 4; ++m)
            #pragma unroll
            for (int n = 0; n < 4; ++n)
                o[8 + 4 * m + n] = u[m] * Amat[4 * m + n] * v[n];
    }
}

// h_pre[t][d] = sum_n H_pre[t][n] * h[t][n][d]
__global__ void __launch_bounds__(256)
hpre_kernel(const float* __restrict__ h, const float* __restrict__ coeffs,
            float* __restrict__ h_pre) {
    int idx = blockIdx.x * 256 + threadIdx.x;      // t*512 + d
    int t = idx >> 9, d = idx & 511;
    const float* hp = h + (size_t)t * NC + d;
    const float* cf = coeffs + (size_t)t * NCOLS;
    h_pre[idx] = cf[0] * hp[0] + cf[1] * hp[DDIM] +
                 cf[2] * hp[2 * DDIM] + cf[3] * hp[3 * DDIM];
}

// h[t][m][d] = H_post[m]*h_out[t][d] + sum_n P[m][n]*h[t][n][d]   (in place)
__global__ void __launch_bounds__(256)
combine_kernel(float* h, const float* __restrict__ h_out,
               const float* __restrict__ coeffs) {
    int idx = blockIdx.x * 256 + threadIdx.x;      // t*512 + d
    int t = idx >> 9, d = idx & 511;
    float* hp = h + (size_t)t * NC + d;
    float h0 = hp[0], h1 = hp[DDIM], h2 = hp[2 * DDIM], h3 = hp[3 * DDIM];
    float ho = h_out[idx];
    const float* cf = coeffs + (size_t)t * NCOLS;
    #pragma unroll
    for (int m = 0; m < 4; ++m) {
        float p0 = cf[8 + 4 * m + 0], p1 = cf[8 + 4 * m + 1];
        float p2 = cf[8 + 4 * m + 2], p3 = cf[8 + 4 * m + 3];
        hp[m * DDIM] = cf[4 + m] * ho + p0 * h0 + p1 * h1 + p2 * h2 + p3 * h3;
    }
}

// out[t][d] = (sum_m h[t][m][d]) * rsqrt(mean_d(y^2)+1e-6) * final_w[d]
__global__ void __launch_bounds__(256)
final_kernel(const float* __restrict__ h, const float* __restrict__ final_w,
             float* __restrict__ out) {
    const int lane = threadIdx.x & 31;
    const int t = blockIdx.x * 8 + (threadIdx.x >> 5);
    const float* hp = h + (size_t)t * NC;
    float y[16];
    float ssq = 0.0f;
    #pragma unroll
    for (int i = 0; i < 16; ++i) {
        int d = i * 32 + lane;
        float v = hp[d] + hp[DDIM + d] + hp[2 * DDIM + d] + hp[3 * DDIM + d];
        y[i] = v;
        ssq += v * v;
    }
    #pragma unroll
    for (int off = 16; off > 0; off >>= 1) ssq += __shfl_xor(ssq, off, 32);
    float scale = rsqrtf(ssq * (1.0f / 512.0f) + 1e-6f);
    #pragma unroll
    for (int i = 0; i < 16; ++i) {
        int d = i * 32 + lane;
        out[(size_t)t * DDIM + d] = y[i] * scale * final_w[d];
    }
}

// ---------------------------------------------------------------------------
extern "C" void kernel_launch(void* const* d_in, const int* in_sizes, int n_in,
                              void* d_out, int out_size, void* d_ws, size_t ws_size,
                              hipStream_t stream) {
    const float* x       = (const float*)d_in[0];
    const float* Wp      = (const float*)d_in[1];
    const float* bp      = (const float*)d_in[2];
    const float* norm_w  = (const float*)d_in[3];
    const float* w       = (const float*)d_in[4];
    const float* alpha   = (const float*)d_in[5];
    const float* beta    = (const float*)d_in[6];
    const float* lin_W   = (const float*)d_in[7];
    const float* lin_b   = (const float*)d_in[8];
    const float* final_w = (const float*)d_in[9];
    float* out = (float*)d_out;

    float* ws     = (float*)d_ws;
    float* h      = ws;                                  // 8192*2048 (64 MB)
    float* h_pre  = h + (size_t)MTOK * NC;               // 8192*512
    float* h_out  = h_pre + (size_t)MTOK * DDIM;         // 8192*512
    float* coeffs = h_out + (size_t)MTOK * DDIM;         // 8192*24

    const dim3 gemm_grid(DDIM / 128, MTOK / 64);
    const int elem_blocks = (MTOK * DDIM) / 256;

    // Input projection (+broadcast into 4 streams of h)
    gemm_wmma_f32<<<gemm_grid, 256, 0, stream>>>(x, Wp, bp, h, INDIM, 0, 1);

    for (int s = 0; s < NSUB; ++s) {
        stats_kernel<<<MTOK / 8, 256, 0, stream>>>(
            h, norm_w + (size_t)s * NC, w + (size_t)s * NC * NCOLS,
            alpha + (size_t)s * 3, beta + (size_t)s * NCOLS, coeffs);
        hpre_kernel<<<elem_blocks, 256, 0, stream>>>(h, coeffs, h_pre);
        gemm_wmma_f32<<<gemm_grid, 256, 0, stream>>>(
            h_pre, lin_W + (size_t)s * DDIM * DDIM, lin_b + (size_t)s * DDIM,
            h_out, DDIM, 1, 0);
        combine_kernel<<<elem_blocks, 256, 0, stream>>>(h, h_out, coeffs);
    }

    final_kernel<<<MTOK / 8, 256, 0, stream>>>(h, final_w, out);
}